// Adversarial_33483565039790
// MI455X (gfx1250) — compile-verified
//
#include <hip/hip_runtime.h>
#include <hip/hip_bf16.h>
#include <stdint.h>

typedef float v4f __attribute__((ext_vector_type(4)));

__device__ __forceinline__ void argsel(float& best, int& bi, float v, int c) {
  // branchless: strict > keeps the FIRST index on ties (matches jnp.argmax)
  const bool t = (v > best);
  best = t ? v : best;
  bi   = t ? c : bi;
}

__device__ __forceinline__ void argcomb(float& best, int& bi, float ov, int oi) {
  // combine two (val,idx) pairs; lower index wins ties
  const bool t = (ov > best) || (ov == best && oi < bi);
  best = t ? ov : best;
  bi   = t ? oi : bi;
}

// ---------------------------------------------------------------------------
// Kernel A: per-sample argmax over 1000 logits, then gather the selected CAM
// channel for all 49 spatial positions and apply the >0.5 threshold.
// One block (8 wave32) per sample; 1000 logits = 250 float4s, so each lane
// does exactly one global_load_b128 and a branchless in-register 4-way argmax.
// ---------------------------------------------------------------------------
__global__ __launch_bounds__(256) void argmax_cam_kernel(
    const float* __restrict__ interm,   // [B, 49, 1000]
    const float* __restrict__ logits,   // [B, 1000]
    float* __restrict__ cam,            // [B, 49]  (workspace)
    int nClasses, int hw)
{
  const int b = blockIdx.x;
  const int n4 = nClasses >> 2;                       // 250
  const v4f* row4 = (const v4f*)(logits + (size_t)b * nClasses);

  float best = -__builtin_inff();
  int   bi   = 0x7FFFFFFF;
  if ((int)threadIdx.x < n4) {
    v4f v = row4[threadIdx.x];
    const int c0 = (int)threadIdx.x * 4;
    best = v.x; bi = c0;
    argsel(best, bi, v.y, c0 + 1);
    argsel(best, bi, v.z, c0 + 2);
    argsel(best, bi, v.w, c0 + 3);
  }

  // wave32 shuffle reduction
  #pragma unroll
  for (int off = 16; off > 0; off >>= 1) {
    float ov = __shfl_xor(best, off, 32);
    int   oi = __shfl_xor(bi,   off, 32);
    argcomb(best, bi, ov, oi);
  }

  __shared__ float sval[8];
  __shared__ int   sidx[8];
  __shared__ int   finalIdx;
  const int wave = threadIdx.x >> 5;
  const int lane = threadIdx.x & 31;
  if (lane == 0) { sval[wave] = best; sidx[wave] = bi; }
  __syncthreads();
  if (threadIdx.x == 0) {
    float fb = sval[0]; int fi = sidx[0];
    #pragma unroll
    for (int w = 1; w < 8; ++w) argcomb(fb, fi, sval[w], sidx[w]);
    finalIdx = fi;
  }
  __syncthreads();

  const int idx = finalIdx;
  if ((int)threadIdx.x < hw) {
    float v = interm[((size_t)b * hw + threadIdx.x) * nClasses + idx];
    cam[(size_t)b * hw + threadIdx.x] = (v > 0.5f) ? v : 0.0f;
  }
}

// ---------------------------------------------------------------------------
// Kernel B: out = vgg - cam_broadcast. Pure HBM stream (global_load_b128 /
// store_b128, nontemporal). The 16 CAM scalars a block needs are staged into
// LDS with the CDNA5 async-to-LDS path (ASYNCcnt) and read back via ds_load.
// Block covers exactly 2048 float4s = 16 channel-groups of 128 float4s.
// ---------------------------------------------------------------------------
__global__ __launch_bounds__(512) void cam_sub_kernel(
    const v4f* __restrict__ vgg,
    const float* __restrict__ cam,
    v4f* __restrict__ out,
    int n4, int camCount)
{
  __shared__ float camLDS[16];

  const int base  = blockIdx.x * (512 * 4);  // first float4 index of this block
  const int gBase = base >> 7;               // first channel-group (16 per block)

  if (threadIdx.x < 16 && (gBase + (int)threadIdx.x) < camCount) {
    const float* gp = cam + gBase + threadIdx.x;
    // Generic pointer to __shared__ carries the LDS byte offset in low 32 bits.
    unsigned loff = (unsigned)(uintptr_t)(&camLDS[threadIdx.x]);
    asm volatile("global_load_async_to_lds_b32 %0, %1, off"
                 :: "v"(loff), "v"(gp) : "memory");
  }
  asm volatile("s_wait_asynccnt 0" ::: "memory");
  __syncthreads();

  #pragma unroll
  for (int k = 0; k < 4; ++k) {
    const int j = base + (int)threadIdx.x + k * 512;
    if (j < n4) {
      v4f v = __builtin_nontemporal_load(&vgg[j]);
      float c = camLDS[(j >> 7) & 15];
      v4f r = v - c;
      __builtin_nontemporal_store(r, &out[j]);
    }
  }
}

// ---------------------------------------------------------------------------
extern "C" void kernel_launch(void* const* d_in, const int* in_sizes, int n_in,
                              void* d_out, int out_size, void* d_ws, size_t ws_size,
                              hipStream_t stream) {
  const float* vgg    = (const float*)d_in[0];  // [256,7,7,512]
  const float* interm = (const float*)d_in[1];  // [256,7,7,1000]
  const float* logits = (const float*)d_in[2];  // [256,1000]
  float* out = (float*)d_out;

  const int nClasses = 1000;
  const int hw       = 49;
  const int B        = in_sizes[2] / nClasses;   // 256
  const int camCount = B * hw;                   // 12544 floats in workspace

  float* cam = (float*)d_ws;

  argmax_cam_kernel<<<B, 256, 0, stream>>>(interm, logits, cam, nClasses, hw);

  const int n4     = out_size / 4;               // 1,605,632 float4s
  const int blocks = (n4 + 2047) / 2048;         // 784
  cam_sub_kernel<<<blocks, 512, 0, stream>>>((const v4f*)vgg, cam, (v4f*)out,
                                             n4, camCount);
}